// LinearInterpolation_20907900797406
// MI455X (gfx1250) — compile-verified
//
#include <hip/hip_runtime.h>

// CDNA5 / gfx1250, wave32. Memory-bound lerp recast as out = W(48x13,const) x V(13x2)
// per batch via chained V_WMMA_F32_16X16X4_F32. W^T (B operand) is grid-constant:
// built once per block by wave 0 into LDS, reloaded by all waves with ds_load_b128.
// Wave-uniform values (wave id, batch base) are forced into SGPRs via readfirstlane
// so stores are unguarded saddr+imm forms on the hot path.

typedef float v2f __attribute__((ext_vector_type(2)));
typedef float v8f __attribute__((ext_vector_type(8)));

#define NKNOTS 13
#define MPOS   48      // idx[last] - idx[0] positions
#define PTILES 3       // 48 / 16
#define BPW    8       // batches per wave: 16 M-rows = 8 batches x 2 channels

__global__ __launch_bounds__(256)
void LinearInterpolation_wmma_kernel(const int* __restrict__ idx,
                                     const float* __restrict__ value,
                                     float* __restrict__ out,
                                     int nbatch)
{
    // B layout in LDS: Bs[pt][lane][2*i+j]  (8 dwords per lane per tile)
    __shared__ float Bs[PTILES][32][8];

    const int lane = threadIdx.x & 31;
    const int lo16 = lane & 15;               // M-row (A) or N-column (B/D)
    const int hi   = lane >> 4;               // half-wave selector (K offset +2)

    // Wave id as a compiler-provable scalar -> b0 lives in an SGPR.
    const int wave = __builtin_amdgcn_readfirstlane(threadIdx.x >> 5);

    // ---------- wave 0: build W^T tiles once per block (wave-uniform branch) ------
    if (wave == 0) {
        int kx[NKNOTS];
#pragma unroll
        for (int j = 0; j < NKNOTS; ++j) kx[j] = idx[j];
#pragma unroll
        for (int pt = 0; pt < PTILES; ++pt) {
            const int pv = kx[0] + 1 + pt * 16 + lo16;    // interpolation position
            int seg = 0;                                  // last j with kx[j] < pv
#pragma unroll
            for (int j = 1; j < NKNOTS - 1; ++j)
                if (kx[j] < pv) seg = j;
            const float x0 = (float)kx[seg];
            const float x1 = (float)kx[seg + 1];
            const float t  = ((float)pv - x0) / (x1 - x0);
#pragma unroll
            for (int i = 0; i < 4; ++i) {
#pragma unroll
                for (int j = 0; j < 2; ++j) {
                    const int K = 4 * i + j + 2 * hi;
                    float w = 0.0f;
                    if (K == seg)     w = 1.0f - t;
                    if (K == seg + 1) w = t;
                    Bs[pt][lane][2 * i + j] = w;
                }
            }
        }
    }
    __syncthreads();

    const int b0 = (blockIdx.x * 8 + wave) * BPW;   // scalar batch base

    // ---------------- A: value rows. M = 8*c + b_local, K = knot index -------------
    // 32-bit A 16x4 layout: VGPRj holds K = 4i+j (lanes 0-15), K = 4i+j+2 (lanes 16-31)
    const int ch = lo16 >> 3;                 // channel 0/1
    const int bl = lo16 & 7;                  // local batch 0..7
    int bld = b0 + bl; if (bld > nbatch - 1) bld = nbatch - 1;   // tail clamp (2 VALU)
    const float* vbase = value + (size_t)bld * (NKNOTS * 2) + ch;

    v2f A[4];
#pragma unroll
    for (int i = 0; i < 4; ++i) {
#pragma unroll
        for (int j = 0; j < 2; ++j) {
            const int K  = 4 * i + j + 2 * hi;
            const int Kc = (K < NKNOTS) ? K : (NKNOTS - 1);   // in-bounds address
            float v = __builtin_nontemporal_load(&vbase[Kc * 2]); // unconditional load
            A[i][j] = (K < NKNOTS) ? v : 0.0f;                // zero pad rows 13..15
        }
    }

    // ---------------- B: reload W^T tiles from LDS (2x ds_load_b128 per tile) -----
    v2f B[PTILES][4];
#pragma unroll
    for (int pt = 0; pt < PTILES; ++pt) {
        const float4 blo = *(const float4*)&Bs[pt][lane][0];
        const float4 bhi = *(const float4*)&Bs[pt][lane][4];
        B[pt][0] = v2f{blo.x, blo.y};
        B[pt][1] = v2f{blo.z, blo.w};
        B[pt][2] = v2f{bhi.x, bhi.y};
        B[pt][3] = v2f{bhi.z, bhi.w};
    }

    // ---------------- D = A x B, K=16 via 4 chained K=4 WMMAs ----------------------
    v8f D[PTILES];
#pragma unroll
    for (int pt = 0; pt < PTILES; ++pt) {
        v8f acc = {};
#pragma unroll
        for (int i = 0; i < 4; ++i) {
            // (neg_a, A, neg_b, B, c_mod, C, reuse_a, reuse_b)
            acc = __builtin_amdgcn_wmma_f32_16x16x4_f32(
                false, A[i], false, B[pt][i], (short)0, acc, false, false);
        }
        D[pt] = acc;
    }

    // ---------------- Stores: one fully dense 128B block per instruction ----------
    // D VGPR r: lanes 0-15 = (batch r, ch0, p=lo16), lanes 16-31 = (batch r, ch1, p)
    // -> even/odd 4B slots of one contiguous 128B span. Write-once => non-temporal.
    float* obase = out + (size_t)b0 * (MPOS * 2);   // scalar base (SGPR)
    const int voff = lo16 * 2 + hi;                 // per-lane dword offset
    if (b0 + BPW <= nbatch) {                       // scalar branch: full fast path
#pragma unroll
        for (int pt = 0; pt < PTILES; ++pt) {
#pragma unroll
            for (int r = 0; r < 8; ++r) {
                __builtin_nontemporal_store(D[pt][r],
                                            obase + voff + r * (MPOS * 2) + pt * 32);
            }
        }
    } else {                                        // cold tail path (never for 262144)
#pragma unroll
        for (int pt = 0; pt < PTILES; ++pt) {
#pragma unroll
            for (int r = 0; r < 8; ++r) {
                if (b0 + r < nbatch) {
                    __builtin_nontemporal_store(D[pt][r],
                                                obase + voff + r * (MPOS * 2) + pt * 32);
                }
            }
        }
    }
}

extern "C" void kernel_launch(void* const* d_in, const int* in_sizes, int n_in,
                              void* d_out, int out_size, void* d_ws, size_t ws_size,
                              hipStream_t stream)
{
    const int*   idx   = (const int*)d_in[0];     // 13 int32 knots
    const float* value = (const float*)d_in[1];   // [BATCH, 13, 2] f32
    float*       out   = (float*)d_out;           // [BATCH, 48, 2] f32

    const int nbatch = in_sizes[1] / (NKNOTS * 2);
    const int waves  = (nbatch + BPW - 1) / BPW;  // one wave per 8 batches
    const int blocks = (waves + 7) / 8;           // 8 waves (256 thr) per block

    LinearInterpolation_wmma_kernel<<<blocks, 256, 0, stream>>>(idx, value, out, nbatch);
}